// Q_NetWork_10737418240259
// MI455X (gfx1250) — compile-verified
//
#include <hip/hip_runtime.h>
#include <math.h>

#define T_LEN   16384
#define NITEMS  1000
#define IEMB    32
#define THID    32      // time LSTM hidden
#define SHID    40      // sub LSTM hidden
#define TGATE   128     // 4*THID
#define SGATE   160     // 4*SHID
#define KPAD    12      // time-LSTM input dim 10 padded to multiple of 4
#define FUSE_DIM 216

typedef float v2f __attribute__((ext_vector_type(2)));
typedef float v8f __attribute__((ext_vector_type(8)));

__device__ __forceinline__ float sigmoidf_(float x) {
    return 1.0f / (1.0f + __expf(-x));
}
__device__ __forceinline__ float tanhf_(float x) {
    float e = __expf(-2.0f * x);
    return (1.0f - e) / (1.0f + e);
}

// ---------------------------------------------------------------------------
// Phase 0: zero-pad time_Wih [128 x 10] -> Wih_pad [128 x 12] so the WMMA GEMM
// can load B unconditionally (no divergent exec-masked loads in the hot loop).
// ---------------------------------------------------------------------------
__global__ void k_pad_wih(const float* __restrict__ Wih,   // [128 x 10]
                          float* __restrict__ Wih_pad) {   // [128 x 12]
    int idx = blockIdx.x * blockDim.x + threadIdx.x;       // 0 .. 128*12-1
    if (idx >= TGATE * KPAD) return;
    int g = idx / KPAD, k = idx % KPAD;
    Wih_pad[idx] = (k < 10) ? Wih[g * 10 + k] : 0.f;
}

// ---------------------------------------------------------------------------
// Phase 1: embedding gather + per-type projection -> xs[T x 12] (cols 10,11 = 0)
// ---------------------------------------------------------------------------
__global__ void k_embed_project(const int* __restrict__ seq_items,
                                const int* __restrict__ seq_types,
                                const float* __restrict__ dwell,
                                const float* __restrict__ item_emb,      // [32 x 1000]
                                const float* __restrict__ click_proj,    // [9 x 32]
                                const float* __restrict__ purchase_proj,
                                const float* __restrict__ skip_proj,
                                float* __restrict__ xs) {                // [T x 12]
    int t = blockIdx.x * blockDim.x + threadIdx.x;
    if (t >= T_LEN) return;
    int item = seq_items[t];
    int ty   = seq_types[t];
    const float* P = (ty == 0) ? click_proj : (ty == 1) ? purchase_proj : skip_proj;
    float e[IEMB];
#pragma unroll
    for (int j = 0; j < IEMB; ++j) e[j] = item_emb[j * NITEMS + item];
    float* xr = xs + (size_t)t * KPAD;
    xr[0] = dwell[t];
#pragma unroll
    for (int r = 0; r < 9; ++r) {
        float acc = 0.f;
#pragma unroll
        for (int j = 0; j < IEMB; ++j) acc += P[r * IEMB + j] * e[j];
        xr[1 + r] = acc;
    }
    xr[10] = 0.f;
    xr[11] = 0.f;
}

// ---------------------------------------------------------------------------
// Phase 2: Xg[T x 128] = xs[T x 12pad] @ Wih_pad.T + (bih+bhh), WMMA f32 16x16x4
// One wave per 16x16 output tile. Grid (T/16, 128/16), block 32.
// ---------------------------------------------------------------------------
__global__ void __launch_bounds__(32)
k_gemm_xg(const float* __restrict__ xs,       // [T x 12]
          const float* __restrict__ Wih_pad,  // [128 x 12]
          const float* __restrict__ bih,
          const float* __restrict__ bhh,
          float* __restrict__ Xg) {           // [T x 128]
    int lane = threadIdx.x;
    int half = lane >> 4;        // 0 or 1
    int l15  = lane & 15;
    int m0 = blockIdx.x * 16;
    int n0 = blockIdx.y * 16;
    int row = m0 + l15;          // A matrix row (M = t)
    int col = n0 + l15;          // B matrix col (N = gate)
    v8f acc = {};
#pragma unroll
    for (int kc = 0; kc < 3; ++kc) {
        int k0 = kc * 4 + half * 2;       // ISA 16x4 f32 layout: [M=L%16, K=v+2*(L/16)]
        v2f a, b;
        a.x = xs[(size_t)row * KPAD + k0];
        a.y = xs[(size_t)row * KPAD + k0 + 1];
        b.x = Wih_pad[col * KPAD + k0];
        b.y = Wih_pad[col * KPAD + k0 + 1];
        acc = __builtin_amdgcn_wmma_f32_16x16x4_f32(false, a, false, b,
                                                    (short)0, acc, false, false);
    }
    float bias = bih[col] + bhh[col];
#pragma unroll
    for (int v = 0; v < 8; ++v) {
        int m = m0 + v + half * 8;        // C/D layout: VGPR v -> M = v + 8*(L/16)
        Xg[(size_t)m * TGATE + col] = acc[v] + bias;
    }
}

// ---------------------------------------------------------------------------
// Phase 3: time LSTM recurrence (the only truly sequential part for ys).
// Single wave; Whh preloaded into LDS transposed (conflict-free); h in LDS.
// Prefetch next timestep's gate row to hide the only global load latency.
// ---------------------------------------------------------------------------
__global__ void __launch_bounds__(32)
k_time_lstm(const float* __restrict__ Xg,    // [T x 128]
            const float* __restrict__ Whh,   // [128 x 32]
            float* __restrict__ ys) {        // [T x 32]
    __shared__ float wsh[THID * TGATE];      // [k][g] transposed, 16 KB
    __shared__ float hs[THID];
    int j = threadIdx.x;
    for (int idx = j; idx < TGATE * THID; idx += 32) {
        int g = idx / THID, k = idx % THID;
        wsh[k * TGATE + g] = Whh[idx];
    }
    hs[j] = 0.f;
    float c = 0.f;
    __syncthreads();
    for (int t = 0; t < T_LEN; ++t) {
        const float* base = Xg + (size_t)t * TGATE;
        if (t + 1 < T_LEN)
            __builtin_prefetch(base + TGATE + j * 4, 0, 0);   // next row -> caches
        float gi = base[j], gf = base[THID + j];
        float gg = base[2 * THID + j], go = base[3 * THID + j];
#pragma unroll 8
        for (int k = 0; k < THID; ++k) {
            float hk = hs[k];                       // LDS broadcast read
            const float* w = wsh + k * TGATE;       // lanes hit distinct banks
            gi += w[j] * hk;
            gf += w[THID + j] * hk;
            gg += w[2 * THID + j] * hk;
            go += w[3 * THID + j] * hk;
        }
        float i = sigmoidf_(gi), f = sigmoidf_(gf);
        float g = tanhf_(gg),    o = sigmoidf_(go);
        c = f * c + i * g;
        float h = o * tanhf_(c);
        __syncthreads();
        hs[j] = h;
        ys[(size_t)t * THID + j] = h;
        __syncthreads();
    }
}

// ---------------------------------------------------------------------------
// Phase 4: G[z][T x 160] = ys[T x 32] @ Wih_z.T + bias_z, WMMA f32 16x16x4
// Grid (T/16, 160/16, 3), block 32.
// ---------------------------------------------------------------------------
__global__ void __launch_bounds__(32)
k_gemm_g(const float* __restrict__ ys,       // [T x 32]
         const float* __restrict__ pWih, const float* __restrict__ pbih, const float* __restrict__ pbhh,
         const float* __restrict__ cWih, const float* __restrict__ cbih, const float* __restrict__ cbhh,
         const float* __restrict__ sWih, const float* __restrict__ sbih, const float* __restrict__ sbhh,
         float* __restrict__ G) {            // [3 x T x 160]
    int z = blockIdx.z;
    const float* Wih = (z == 0) ? pWih : (z == 1) ? cWih : sWih;
    const float* bih = (z == 0) ? pbih : (z == 1) ? cbih : sbih;
    const float* bhh = (z == 0) ? pbhh : (z == 1) ? cbhh : sbhh;
    float* Gz = G + (size_t)z * T_LEN * SGATE;
    int lane = threadIdx.x;
    int half = lane >> 4, l15 = lane & 15;
    int m0 = blockIdx.x * 16, n0 = blockIdx.y * 16;
    int row = m0 + l15, col = n0 + l15;
    v8f acc = {};
#pragma unroll
    for (int kc = 0; kc < 8; ++kc) {
        int k0 = kc * 4 + half * 2;
        v2f a, b;
        a.x = ys[(size_t)row * THID + k0];
        a.y = ys[(size_t)row * THID + k0 + 1];
        b.x = Wih[col * THID + k0];
        b.y = Wih[col * THID + k0 + 1];
        acc = __builtin_amdgcn_wmma_f32_16x16x4_f32(false, a, false, b,
                                                    (short)0, acc, false, false);
    }
    float bias = bih[col] + bhh[col];
#pragma unroll
    for (int v = 0; v < 8; ++v) {
        Gz[(size_t)(m0 + v + half * 8) * SGATE + col] = acc[v] + bias;
    }
}

// ---------------------------------------------------------------------------
// Phase 5: three masked LSTMs, one block each (z=0:p mask==1, 1:c mask==0, 2:s mask==2).
// Skips ~2/3 of timesteps each; input gates precomputed in G.
// ---------------------------------------------------------------------------
__global__ void __launch_bounds__(64)
k_sub_lstm(const float* __restrict__ G,          // [3 x T x 160]
           const int* __restrict__ seq_types,
           const float* __restrict__ pWhh,       // [160 x 40]
           const float* __restrict__ cWhh,
           const float* __restrict__ sWhh,
           float* __restrict__ Hout) {           // [3 x 40] order p,c,s
    int z = blockIdx.x;
    const float* Whh = (z == 0) ? pWhh : (z == 1) ? cWhh : sWhh;
    const int mtype  = (z == 0) ? 1    : (z == 1) ? 0    : 2;
    const float* Gz = G + (size_t)z * T_LEN * SGATE;
    __shared__ float wsh[SHID * SGATE];          // [k][g] transposed, 25.6 KB
    __shared__ float hs[SHID];
    int j = threadIdx.x;
    for (int idx = j; idx < SGATE * SHID; idx += 64) {
        int g = idx / SHID, k = idx % SHID;
        wsh[k * SGATE + g] = Whh[idx];
    }
    if (j < SHID) hs[j] = 0.f;
    float c = 0.f;
    __syncthreads();
    for (int t = 0; t < T_LEN; ++t) {
        if (seq_types[t] != mtype) continue;     // uniform across block: barrier-safe
        float h = 0.f;
        if (j < SHID) {
            const float* base = Gz + (size_t)t * SGATE;
            if (t + 1 < T_LEN)
                __builtin_prefetch(base + SGATE + j * 4, 0, 0);
            float gi = base[j], gf = base[SHID + j];
            float gg = base[2 * SHID + j], go = base[3 * SHID + j];
#pragma unroll 8
            for (int k = 0; k < SHID; ++k) {
                float hk = hs[k];
                const float* w = wsh + k * SGATE;
                gi += w[j] * hk;
                gf += w[SHID + j] * hk;
                gg += w[2 * SHID + j] * hk;
                go += w[3 * SHID + j] * hk;
            }
            float ii = sigmoidf_(gi), ff = sigmoidf_(gf);
            float g2 = tanhf_(gg),    oo = sigmoidf_(go);
            c = ff * c + ii * g2;
            h = oo * tanhf_(c);
        }
        __syncthreads();
        if (j < SHID) hs[j] = h;
        __syncthreads();
    }
    if (j < SHID) Hout[z * SHID + j] = hs[j];
}

// ---------------------------------------------------------------------------
// Phase 6: fusion MLP  out = out2_W @ relu(out1_W @ fusion + out1_b) + out2_b
// fusion = [next_e(32), user_e(32), H_r(32), H_s(40), H_c(40), H_p(40)]
// ---------------------------------------------------------------------------
__global__ void __launch_bounds__(64)
k_fusion(const int* __restrict__ user,
         const int* __restrict__ nextitem,
         const float* __restrict__ item_emb,
         const float* __restrict__ user_emb,
         const float* __restrict__ ys,
         const float* __restrict__ Hout,         // [3 x 40] p,c,s
         const float* __restrict__ out1_W,       // [16 x 216]
         const float* __restrict__ out1_b,
         const float* __restrict__ out2_W,       // [1 x 16]
         const float* __restrict__ out2_b,
         float* __restrict__ out) {
    __shared__ float fus[FUSE_DIM];
    __shared__ float r16[16];
    int j = threadIdx.x;
    int u = user[0], ni = nextitem[0];
    for (int idx = j; idx < FUSE_DIM; idx += 64) {
        float v;
        if (idx < 32)       v = item_emb[idx * NITEMS + ni];
        else if (idx < 64)  v = user_emb[(idx - 32) * NITEMS + u];
        else if (idx < 96)  v = ys[(size_t)(T_LEN - 1) * THID + (idx - 64)];     // H_r
        else if (idx < 136) v = Hout[2 * SHID + (idx - 96)];                      // H_s
        else if (idx < 176) v = Hout[1 * SHID + (idx - 136)];                     // H_c
        else                v = Hout[0 * SHID + (idx - 176)];                     // H_p
        fus[idx] = v;
    }
    __syncthreads();
    if (j < 16) {
        float acc = out1_b[j];
        for (int k = 0; k < FUSE_DIM; ++k) acc += out1_W[j * FUSE_DIM + k] * fus[k];
        r16[j] = fmaxf(acc, 0.f);
    }
    __syncthreads();
    if (j == 0) {
        float acc = out2_b[0];
#pragma unroll
        for (int k = 0; k < 16; ++k) acc += out2_W[k] * r16[k];
        out[0] = acc;
    }
}

// ---------------------------------------------------------------------------
extern "C" void kernel_launch(void* const* d_in, const int* in_sizes, int n_in,
                              void* d_out, int out_size, void* d_ws, size_t ws_size,
                              hipStream_t stream) {
    const int*   user          = (const int*)  d_in[0];
    const int*   nextitem      = (const int*)  d_in[1];
    const int*   seq_items     = (const int*)  d_in[2];
    const int*   seq_types     = (const int*)  d_in[3];
    const float* dwell         = (const float*)d_in[4];
    const float* item_emb      = (const float*)d_in[5];
    const float* user_emb      = (const float*)d_in[6];
    const float* click_proj    = (const float*)d_in[7];
    const float* purchase_proj = (const float*)d_in[8];
    const float* skip_proj     = (const float*)d_in[9];
    const float* time_Wih      = (const float*)d_in[10];
    const float* time_Whh      = (const float*)d_in[11];
    const float* time_bih      = (const float*)d_in[12];
    const float* time_bhh      = (const float*)d_in[13];
    const float* p_Wih         = (const float*)d_in[14];
    const float* p_Whh         = (const float*)d_in[15];
    const float* p_bih         = (const float*)d_in[16];
    const float* p_bhh         = (const float*)d_in[17];
    const float* c_Wih         = (const float*)d_in[18];
    const float* c_Whh         = (const float*)d_in[19];
    const float* c_bih         = (const float*)d_in[20];
    const float* c_bhh         = (const float*)d_in[21];
    const float* s_Wih         = (const float*)d_in[22];
    const float* s_Whh         = (const float*)d_in[23];
    const float* s_bih         = (const float*)d_in[24];
    const float* s_bhh         = (const float*)d_in[25];
    const float* out1_W        = (const float*)d_in[26];
    const float* out1_b        = (const float*)d_in[27];
    const float* out2_W        = (const float*)d_in[28];
    const float* out2_b        = (const float*)d_in[29];

    // workspace layout (floats): xs | Xg | ys | G | Hout | Wih_pad  (~43 MB)
    float* ws      = (float*)d_ws;
    float* xs      = ws;                                   // T*12
    float* Xg      = xs + (size_t)T_LEN * KPAD;            // T*128
    float* ysb     = Xg + (size_t)T_LEN * TGATE;           // T*32
    float* G       = ysb + (size_t)T_LEN * THID;           // 3*T*160
    float* Hout    = G + 3 * (size_t)T_LEN * SGATE;        // 120 (+8 pad)
    float* Wih_pad = Hout + 128;                           // 128*12

    k_pad_wih<<<(TGATE * KPAD + 255) / 256, 256, 0, stream>>>(time_Wih, Wih_pad);

    k_embed_project<<<T_LEN / 256, 256, 0, stream>>>(
        seq_items, seq_types, dwell, item_emb, click_proj, purchase_proj, skip_proj, xs);

    k_gemm_xg<<<dim3(T_LEN / 16, TGATE / 16), 32, 0, stream>>>(
        xs, Wih_pad, time_bih, time_bhh, Xg);

    k_time_lstm<<<1, 32, 0, stream>>>(Xg, time_Whh, ysb);

    k_gemm_g<<<dim3(T_LEN / 16, SGATE / 16, 3), 32, 0, stream>>>(
        ysb, p_Wih, p_bih, p_bhh, c_Wih, c_bih, c_bhh, s_Wih, s_bih, s_bhh, G);

    k_sub_lstm<<<3, 64, 0, stream>>>(G, seq_types, p_Whh, c_Whh, s_Whh, Hout);

    k_fusion<<<1, 64, 0, stream>>>(user, nextitem, item_emb, user_emb, ysb, Hout,
                                   out1_W, out1_b, out2_W, out2_b, (float*)d_out);
}